// CRF_60919816126950
// MI455X (gfx1250) — compile-verified
//
#include <hip/hip_runtime.h>
#include <hip/hip_bf16.h>
#include <math.h>

// Problem dims (fixed by the reference)
#define LL 1024
#define BB 256
#define CC 96
#define HH 768

typedef float v2f __attribute__((ext_vector_type(2)));
typedef float v8f __attribute__((ext_vector_type(8)));

// ---------------------------------------------------------------------------
// Block-wide reductions (wave32: shfl within wave, LDS across waves).
// All threads of the block must participate; inactive lanes pass identity.
// ---------------------------------------------------------------------------
__device__ __forceinline__ float block_reduce(float v, float* red, int tid,
                                              int nthreads, bool do_max) {
#pragma unroll
  for (int o = 16; o > 0; o >>= 1) {
    float o2 = __shfl_xor(v, o, 32);
    v = do_max ? fmaxf(v, o2) : (v + o2);
  }
  if ((tid & 31) == 0) red[tid >> 5] = v;
  __syncthreads();
  int nw = nthreads >> 5;
  float r = red[0];
  for (int i = 1; i < nw; ++i) r = do_max ? fmaxf(r, red[i]) : (r + red[i]);
  __syncthreads();
  return r;
}

// ---------------------------------------------------------------------------
// GEMM: out[M x N] = A[M x K] @ Bsrc[N x K]^T + bias[N]
// fp32 WMMA 16x16x4. Block = 128 threads = 4 waves; each wave owns one
// 16-row M tile and JT 16-col N tiles (compile-time -> single basic block).
// Software-pipelined: fragments for k-group g+1 are loaded before the WMMAs
// of group g execute, so each WMMA group overlaps the next group's loads.
//
// fp32 A-fragment layout (ISA 7.12.2, 16x4): lane l: m = l%16,
//   koff = 2*(l/16); a.x = A[m][k+koff], a.y = A[m][k+koff+1].
// B (4x16, K x N) mirrors it with n = l%16 and B[k][n] = Bsrc[n][k],
// so both fragments are contiguous float2 loads from row-major sources.
// C/D: VGPR v, half = l/16: out row = v + 8*half, col = l%16.
// Launch requires: N == gridDim.x * JT * 16, M == gridDim.y * 64,
// and K % 8 == 0, K >= 16.
// ---------------------------------------------------------------------------
template <int JT>
__global__ void gemm_wmma_f32(const float* __restrict__ A,
                              const float* __restrict__ Bsrc,
                              const float* __restrict__ bias,
                              float* __restrict__ out,
                              int N, int K) {
  const int lane  = threadIdx.x & 31;
  const int wave  = threadIdx.x >> 5;
  const int mbase = (blockIdx.y * 4 + wave) * 16;
  const int nb0   = blockIdx.x * (JT * 16);
  const int mrow  = lane & 15;          // m (and n) index within tile
  const int half  = lane >> 4;          // 0 or 1
  const int koff  = half << 1;          // 0 or 2

  v8f zero = {};
  v8f acc[JT];
#pragma unroll
  for (int j = 0; j < JT; ++j) acc[j] = zero;

  const float* Arow = A + (size_t)(mbase + mrow) * K + koff;
  const float* Brow[JT];
#pragma unroll
  for (int j = 0; j < JT; ++j)
    Brow[j] = Bsrc + (size_t)(nb0 + j * 16 + mrow) * K + koff;

  // ---- software pipeline: buffer A holds group k, buffer B holds k+4 ----
  v2f aA = *reinterpret_cast<const v2f*>(Arow);
  v2f bA[JT];
#pragma unroll
  for (int j = 0; j < JT; ++j) bA[j] = *reinterpret_cast<const v2f*>(Brow[j]);

  v2f aB;
  v2f bB[JT];
  int k = 0;
  for (; k < K - 8; k += 8) {
    // prefetch group k+4
    aB = *reinterpret_cast<const v2f*>(Arow + k + 4);
#pragma unroll
    for (int j = 0; j < JT; ++j)
      bB[j] = *reinterpret_cast<const v2f*>(Brow[j] + k + 4);
    // compute group k (loads for k+4 in flight)
#pragma unroll
    for (int j = 0; j < JT; ++j)
      acc[j] = __builtin_amdgcn_wmma_f32_16x16x4_f32(
          false, aA, false, bA[j], (short)0, acc[j], false, false);
    // prefetch group k+8
    aA = *reinterpret_cast<const v2f*>(Arow + k + 8);
#pragma unroll
    for (int j = 0; j < JT; ++j)
      bA[j] = *reinterpret_cast<const v2f*>(Brow[j] + k + 8);
    // compute group k+4 (loads for k+8 in flight)
#pragma unroll
    for (int j = 0; j < JT; ++j)
      acc[j] = __builtin_amdgcn_wmma_f32_16x16x4_f32(
          false, aB, false, bB[j], (short)0, acc[j], false, false);
  }
  // epilogue: k == K-8; groups k (in A-buffers) and k+4
  aB = *reinterpret_cast<const v2f*>(Arow + k + 4);
#pragma unroll
  for (int j = 0; j < JT; ++j)
    bB[j] = *reinterpret_cast<const v2f*>(Brow[j] + k + 4);
#pragma unroll
  for (int j = 0; j < JT; ++j)
    acc[j] = __builtin_amdgcn_wmma_f32_16x16x4_f32(
        false, aA, false, bA[j], (short)0, acc[j], false, false);
#pragma unroll
  for (int j = 0; j < JT; ++j)
    acc[j] = __builtin_amdgcn_wmma_f32_16x16x4_f32(
        false, aB, false, bB[j], (short)0, acc[j], false, false);

#pragma unroll
  for (int j = 0; j < JT; ++j) {
    int ncol = nb0 + j * 16 + mrow;
    float bv = bias[ncol];
#pragma unroll
    for (int v = 0; v < 8; ++v) {
      int row = mbase + v + 8 * half;
      out[(size_t)row * N + ncol] = acc[j][v] + bv;
    }
  }
}

// ---------------------------------------------------------------------------
// Forward scan (one block per batch element b, 128 threads = 4 waves).
// trans[b] (96x96 = 36 KB) lives in LDS for all 1024 steps.
// P[c][c'] = exp(trans - colmax[c']) precomputed once; each step is then a
// pure FMA matvec: sum[c'] = sum_c exp(alpha[c]-amax) * P[c][c'], and
// alpha'[c'] = log(sum) + amax + colmax[c'] + emit[i,b,c'].
// Shift (amax + colmax) >= true max, so all exponents <= 0 (stable).
// Mask is a prefix (i < lens[b]) -> scan simply stops at len.
// Tail: atomicAdd of logsumexp(alpha + t_end) into accum[0].
// ---------------------------------------------------------------------------
__global__ void crf_scan(const float* __restrict__ emit,
                         const unsigned char* __restrict__ maskb,
                         const float* __restrict__ trans,
                         const float* __restrict__ tstart,
                         const float* __restrict__ tend,
                         float* __restrict__ accum) {
  __shared__ float P[CC * CC];
  __shared__ float eL[CC];
  __shared__ float red[4];

  const int b   = blockIdx.x;
  const int tid = threadIdx.x;   // 0..127

  // Load trans[b] into LDS
  const float* tb = trans + (size_t)b * CC * CC;
  for (int i = tid; i < CC * CC; i += 128) P[i] = tb[i];
  __syncthreads();

  // Column max + exp-domain transform (lane tid owns column tid; the
  // cross-lane LDS reads below are conflict-free: addr = c*96 + tid).
  float colmax = -INFINITY;
  float alpha  = -INFINITY;
  if (tid < CC) {
    for (int c = 0; c < CC; ++c) colmax = fmaxf(colmax, P[c * CC + tid]);
    for (int c = 0; c < CC; ++c) P[c * CC + tid] = __expf(P[c * CC + tid] - colmax);
    alpha = emit[(size_t)b * CC + tid] + tstart[(size_t)b * CC + tid];
  }

  // len_b = sum of mask column b (prefix mask). block_reduce barriers also
  // make the P transform visible to all waves before the matvec below.
  float lf = 0.0f;
  for (int l = tid; l < LL; l += 128)
    lf += maskb[(size_t)l * BB + b] ? 1.0f : 0.0f;
  int len = (int)block_reduce(lf, red, tid, 128, /*max=*/false);

  for (int i = 1; i < len; ++i) {
    float amax = block_reduce(tid < CC ? alpha : -INFINITY, red, tid, 128, true);
    if (tid < CC) eL[tid] = __expf(alpha - amax);
    __syncthreads();
    if (tid < CC) {
      float sum = 0.0f;
#pragma unroll 8
      for (int c = 0; c < CC; ++c)
        sum = fmaf(eL[c], P[c * CC + tid], sum);
      alpha = __logf(sum) + amax + colmax +
              emit[((size_t)i * BB + b) * CC + tid];
    }
    __syncthreads();   // protect eL before next step rewrites it
  }

  // log_z contribution: logsumexp_c(alpha + t_end[b])
  float v  = (tid < CC) ? (alpha + tend[(size_t)b * CC + tid]) : -INFINITY;
  float m  = block_reduce(v, red, tid, 128, true);
  float ex = (tid < CC) ? __expf(v - m) : 0.0f;
  float s  = block_reduce(ex, red, tid, 128, false);
  if (tid == 0) atomicAdd(accum + 0, __logf(s) + m);
}

// ---------------------------------------------------------------------------
// Path score (one block per b, 256 threads striding over L):
//   sum_l mask[l] * ( emit[l,tgt_l] + (l>0) * trans[b,tgt_{l-1},tgt_l] )
//   + t_start[b,tgt_0] + t_end[b,tgt_{len-1}]
// Accumulated into accum[1] via float atomics.
// ---------------------------------------------------------------------------
__global__ void crf_score(const float* __restrict__ emit,
                          const int* __restrict__ target,
                          const unsigned char* __restrict__ maskb,
                          const float* __restrict__ trans,
                          const float* __restrict__ tstart,
                          const float* __restrict__ tend,
                          float* __restrict__ accum) {
  __shared__ float red[8];
  const int b   = blockIdx.x;
  const int tid = threadIdx.x;   // 0..255

  float local = 0.0f, lf = 0.0f;
  for (int l = tid; l < LL; l += 256) {
    if (maskb[(size_t)l * BB + b]) {
      lf += 1.0f;
      int tgt = target[(size_t)l * BB + b];
      float s = emit[((size_t)l * BB + b) * CC + tgt];
      if (l > 0) {
        int prev = target[(size_t)(l - 1) * BB + b];
        s += trans[(size_t)b * CC * CC + (size_t)prev * CC + tgt];
      }
      local += s;
    }
  }
  float total = block_reduce(local, red, tid, 256, false);
  float lenf  = block_reduce(lf,    red, tid, 256, false);
  if (tid == 0) {
    int len = (int)lenf;
    int t0  = target[b];                                  // target[0, b]
    int tl  = target[(size_t)(len - 1) * BB + b];         // last valid tag
    total += tstart[(size_t)b * CC + t0] + tend[(size_t)b * CC + tl];
    atomicAdd(accum + 1, total);
  }
}

__global__ void crf_finalize(const float* __restrict__ accum,
                             float* __restrict__ out) {
  if (blockIdx.x == 0 && threadIdx.x == 0)
    out[0] = (accum[0] - accum[1]) * (1.0f / (float)BB);
}

// ---------------------------------------------------------------------------
// Launch: memset accumulators -> 3 WMMA GEMMs -> scan -> score -> finalize.
// Workspace layout (floats): trans[B*C*C] | t_start[B*C] | t_end[B*C] | accum[2]
// ---------------------------------------------------------------------------
extern "C" void kernel_launch(void* const* d_in, const int* in_sizes, int n_in,
                              void* d_out, int out_size, void* d_ws, size_t ws_size,
                              hipStream_t stream) {
  const float*         emit   = (const float*)d_in[0];
  const float*         hidden = (const float*)d_in[1];
  const int*           target = (const int*)d_in[2];
  const unsigned char* maskb  = (const unsigned char*)d_in[3]; // jnp bool = 1 byte
  const float*         Wt     = (const float*)d_in[4];
  const float*         bt     = (const float*)d_in[5];
  const float*         Ws     = (const float*)d_in[6];
  const float*         bs     = (const float*)d_in[7];
  const float*         We     = (const float*)d_in[8];
  const float*         be     = (const float*)d_in[9];

  float* ws     = (float*)d_ws;
  float* trans  = ws;                                   // B*C*C
  float* tstart = trans + (size_t)BB * CC * CC;         // B*C
  float* tend   = tstart + (size_t)BB * CC;             // B*C
  float* accum  = tend + (size_t)BB * CC;               // 2 floats

  hipMemsetAsync(accum, 0, 2 * sizeof(float), stream);

  // trans = hidden @ Wt^T + bt   (M=256, N=9216, K=768): 144 x 4 blocks,
  // each block = 4 waves x (16M x 64N) tiles.
  dim3 gT((CC * CC) / 64, BB / 64);
  gemm_wmma_f32<4><<<gT, 128, 0, stream>>>(hidden, Wt, bt, trans, CC * CC, HH);
  // t_start / t_end (M=256, N=96 = 6 tiles of 16, K=768)
  dim3 gS(1, BB / 64);
  gemm_wmma_f32<6><<<gS, 128, 0, stream>>>(hidden, Ws, bs, tstart, CC, HH);
  gemm_wmma_f32<6><<<gS, 128, 0, stream>>>(hidden, We, be, tend, CC, HH);

  crf_scan<<<BB, 128, 0, stream>>>(emit, maskb, trans, tstart, tend, accum);
  crf_score<<<BB, 256, 0, stream>>>(emit, target, maskb, trans, tstart, tend, accum);
  crf_finalize<<<1, 32, 0, stream>>>(accum, (float*)d_out);
}